// CapLayer2_6536940225112
// MI455X (gfx1250) — compile-verified
//
#include <hip/hip_runtime.h>
#include <math.h>

typedef float v2f __attribute__((ext_vector_type(2)));
typedef float v8f __attribute__((ext_vector_type(8)));

#define BM 128
#define BN 128
#define BK 32
#define AP (BK + 4)   // A tile K-stride: pad to 36 floats (16B aligned, bank-spread)

// ---------------------------------------------------------------------------
// Kernel 1: pred[b, o, hw] = sum_i W[o,i] * x[b,i,hw] + Wb[o]
//   GEMM: M=2048 (o), K=256 (i), N=16384 (n = b*256 + hw)
//   grid = (128, 16), block = 256 threads (8 waves, 2x4 wave grid)
//   V_WMMA_F32_16X16X4_F32 + double-buffered GLOBAL_LOAD_ASYNC_TO_LDS_B128.
// ---------------------------------------------------------------------------
__global__ __launch_bounds__(256) void caps_gemm_pred_kernel(
    const float* __restrict__ W,    // [2048, 256] row-major
    const float* __restrict__ X,    // [64, 256, 256]  (b, i, hw)
    const float* __restrict__ Wb,   // [2048]
    float* __restrict__ pred)       // [64, 2048, 256] (b, o, hw)
{
    __shared__ float As[2][BM][AP];   // M-major A tile (frag = 1x ds_load_b64)
    __shared__ float Bs[2][BK][BN];   // K-major B tile

    const int t   = threadIdx.x;
    const int n0  = blockIdx.x * BN;       // global column base (0..16383)
    const int m0  = blockIdx.y * BM;       // global row base (0..2047)
    const int bi  = n0 >> 8;               // batch index
    const int hw0 = n0 & 255;              // pixel base within batch

    const int wave = t >> 5;
    const int lane = t & 31;
    const int half = lane >> 4;            // 0: lanes 0-15, 1: lanes 16-31
    const int lm   = lane & 15;
    const int wm   = wave >> 2;            // 0..1  (64-row slab)
    const int wn   = wave & 3;             // 0..3  (32-col slab)

    // per-thread load coordinates (16B granules)
    const int rowA = t >> 3;               // 0..31  (+32 per q)
    const int c4A  = t & 7;                // 0..7
    const int rowB = t >> 5;               // 0..7   (+8 per q)
    const int c4B  = t & 31;               // 0..31

    // Issue one K-tile's async loads (memory -> LDS, tracked by ASYNCcnt)
    auto issue_tile = [&](int k0, int buf) {
        #pragma unroll
        for (int q = 0; q < 4; ++q) {
            int row = rowA + 32 * q;
            unsigned loff = (unsigned)(uintptr_t)&As[buf][row][c4A * 4];
            uint64_t g = (uint64_t)(uintptr_t)(
                W + (size_t)(m0 + row) * 256 + k0 + c4A * 4);
            asm volatile("global_load_async_to_lds_b128 %0, %1, off"
                         :: "v"(loff), "v"(g) : "memory");
        }
        #pragma unroll
        for (int q = 0; q < 4; ++q) {
            int row = rowB + 8 * q;
            unsigned loff = (unsigned)(uintptr_t)&Bs[buf][row][c4B * 4];
            uint64_t g = (uint64_t)(uintptr_t)(
                X + (size_t)bi * 65536 + (size_t)(k0 + row) * 256 + hw0 + c4B * 4);
            asm volatile("global_load_async_to_lds_b128 %0, %1, off"
                         :: "v"(loff), "v"(g) : "memory");
        }
    };

    v8f acc[4][2];
    #pragma unroll
    for (int mi = 0; mi < 4; ++mi)
        #pragma unroll
        for (int ni = 0; ni < 2; ++ni)
            acc[mi][ni] = {};

    issue_tile(0, 0);                       // prologue

    for (int kt = 0; kt < 8; ++kt) {
        const int cur = kt & 1;
        asm volatile("s_wait_asynccnt 0x0" ::: "memory");
        __syncthreads();                    // buf[cur] landed; buf[1-cur] free
        if (kt < 7) issue_tile((kt + 1) * BK, 1 - cur);

        #pragma unroll
        for (int kk = 0; kk < BK; kk += 4) {
            // A fragment 16x4: VGPR0 = K(0|2), VGPR1 = K(1|3) per lane half.
            // M-major LDS layout -> one aligned 8-byte load per fragment.
            v2f a[4];
            #pragma unroll
            for (int mi = 0; mi < 4; ++mi) {
                int rm = wm * 64 + mi * 16 + lm;
                a[mi] = *(const v2f*)&As[cur][rm][kk + 2 * half];
            }
            // B fragment 4x16 (N on lanes, K split by lane half)
            v2f b[2];
            #pragma unroll
            for (int ni = 0; ni < 2; ++ni) {
                int cn = wn * 32 + ni * 16 + lm;
                b[ni].x = Bs[cur][kk + 2 * half + 0][cn];
                b[ni].y = Bs[cur][kk + 2 * half + 1][cn];
            }
            #pragma unroll
            for (int mi = 0; mi < 4; ++mi)
                #pragma unroll
                for (int ni = 0; ni < 2; ++ni)
                    acc[mi][ni] = __builtin_amdgcn_wmma_f32_16x16x4_f32(
                        false, a[mi], false, b[ni],
                        (short)0, acc[mi][ni], false, false);
        }
    }

    // ---- store: C/D layout: VGPR r -> M = r + 8*half, N = lm ----
    #pragma unroll
    for (int mi = 0; mi < 4; ++mi) {
        #pragma unroll
        for (int ni = 0; ni < 2; ++ni) {
            int gn_hw = hw0 + wn * 32 + ni * 16 + lm;
            #pragma unroll
            for (int r = 0; r < 8; ++r) {
                int o = m0 + wm * 64 + mi * 16 + half * 8 + r;
                pred[(size_t)bi * 524288 + (size_t)o * 256 + gn_hw] =
                    acc[mi][ni][r] + Wb[o];
            }
        }
    }
}

// ---------------------------------------------------------------------------
// Kernel 2: dynamic routing, one workgroup per batch. pred stays in L2.
// ---------------------------------------------------------------------------
#define JCAPS 64
#define ICAPS 256
#define D2    32

__global__ __launch_bounds__(256) void caps_routing_kernel(
    const float* __restrict__ b_init,  // [64, 64, 256]
    const float* __restrict__ pred,    // [64, 2048, 256]  (b, j*32+d, i)
    float* __restrict__ b_ws,          // [64, 64, 256]
    float* __restrict__ out)           // [64, 64, 32]
{
    __shared__ float c[JCAPS][ICAPS];
    __shared__ float sv[JCAPS][D2];

    const int bi = blockIdx.x;
    const int t  = threadIdx.x;
    const float* predb = pred + (size_t)bi * 524288;

    for (int it = 0; it < 3; ++it) {
        const float* bsrc = (it == 0) ? (b_init + (size_t)bi * 16384)
                                      : (b_ws + (size_t)bi * 16384);
        // ---- softmax over J for column i = t ----
        {
            float bv[JCAPS];
            float mx = -INFINITY;
            #pragma unroll
            for (int j = 0; j < JCAPS; ++j) {
                bv[j] = bsrc[j * 256 + t];
                mx = fmaxf(mx, bv[j]);
            }
            float sum = 0.f;
            #pragma unroll
            for (int j = 0; j < JCAPS; ++j) {
                float e = __expf(bv[j] - mx);
                bv[j] = e;
                sum += e;
            }
            float inv = 1.f / sum;
            #pragma unroll
            for (int j = 0; j < JCAPS; ++j)
                c[j][t] = bv[j] * inv;
        }
        __syncthreads();

        // ---- s[j][d] = sum_i c[j][i] * pred[(j*32+d), i] ----
        {
            int d  = t & 31;       // lane -> d (uniform j within a wave)
            int jb = t >> 5;
            #pragma unroll
            for (int r = 0; r < 8; ++r) {
                int j = jb + 8 * r;
                const float* pr = predb + (size_t)(j * 32 + d) * 256;
                float a0 = 0.f, a1 = 0.f, a2 = 0.f, a3 = 0.f;
                for (int i = 0; i < 256; i += 4) {
                    float4 p = *(const float4*)(pr + i);
                    a0 += c[j][i + 0] * p.x;
                    a1 += c[j][i + 1] * p.y;
                    a2 += c[j][i + 2] * p.z;
                    a3 += c[j][i + 3] * p.w;
                }
                sv[j][d] = (a0 + a1) + (a2 + a3);
            }
        }
        __syncthreads();

        // ---- v = squash(s), in place ----
        if (t < JCAPS) {
            float sq = 0.f;
            #pragma unroll
            for (int d = 0; d < D2; ++d) { float x = sv[t][d]; sq += x * x; }
            float scale = (sq / (1.f + sq)) * rsqrtf(sq + 1e-8f);
            #pragma unroll
            for (int d = 0; d < D2; ++d) sv[t][d] *= scale;
        }
        __syncthreads();

        // ---- b += <v, pred>  (dead in last iteration -> skip) ----
        if (it < 2) {
            // thread t handles column i = t for all j (coalesced over lanes)
            for (int j = 0; j < JCAPS; ++j) {
                float acc = 0.f;
                #pragma unroll
                for (int d = 0; d < D2; ++d)
                    acc += sv[j][d] * predb[(size_t)(j * 32 + d) * 256 + t];
                b_ws[(size_t)bi * 16384 + j * 256 + t] = bsrc[j * 256 + t] + acc;
            }
            __syncthreads();
        }
    }

    // ---- write v: out[bi, j, d] ----
    #pragma unroll
    for (int q = 0; q < 8; ++q) {
        int p = t + 256 * q;                       // 0..2047 = j*32+d
        out[(size_t)bi * 2048 + p] = ((const float*)sv)[p];
    }
}

// ---------------------------------------------------------------------------
// Launch
// ---------------------------------------------------------------------------
extern "C" void kernel_launch(void* const* d_in, const int* in_sizes, int n_in,
                              void* d_out, int out_size, void* d_ws, size_t ws_size,
                              hipStream_t stream) {
    const float* x      = (const float*)d_in[0];   // [64,256,16,16]
    const float* b_init = (const float*)d_in[1];   // [64,64,256]
    const float* W      = (const float*)d_in[2];   // [2048,256]
    const float* Wb     = (const float*)d_in[3];   // [2048]
    float* out = (float*)d_out;                    // [64,64,32]

    // workspace: pred (134,217,728 B) | b logits (4,194,304 B)
    float* pred = (float*)d_ws;
    float* b_ws = (float*)((char*)d_ws + (size_t)134217728);

    dim3 ggrid(16384 / BN, 2048 / BM);             // (128, 16)
    caps_gemm_pred_kernel<<<ggrid, 256, 0, stream>>>(W, x, Wb, pred);

    caps_routing_kernel<<<64, 256, 0, stream>>>(b_init, pred, b_ws, out);
}